// FCOSOutputs_49151605735964
// MI455X (gfx1250) — compile-verified
//
#include <hip/hip_runtime.h>
#include <hip/hip_bf16.h>

// FCOS target assignment for MI455X (gfx1250, wave32).
//
// Core trick: the outer-difference regression tensor
//   l(i,j)=xs_i-x1_j, t=ys_i-y1_j, r=x2_j-xs_i, b=y2_j-ys_i
// is a rank-2 GEMM: A = [xs, ys, 1, 0] (16x4 f32), B column per component,
// computed with V_WMMA_F32_16X16X4_F32 (exact: all products are *1/*0, one
// nonzero add -> bitwise equal to the f32 reference subtraction).
//
// One wave handles a 16-location x 16-box tile; 7 box tiles cover M=100
// (padded to 112 with sentinel boxes that can never win the argmin).
// Argmin over boxes: branchless per-lane running min across tiles, then
// shfl_xor reduction over each 16-lane half (tie-break on lower box index,
// matching jnp.argmin first-min semantics), then per-wave LDS transpose so
// lanes 0-15 each own one location for label/reg-target computation.
//
// All predicates use bitwise & / explicit selects so the compiler emits
// v_cmp + v_cndmask dataflow instead of EXEC-mask branches.

#define INF_VAL      100000000.0f
#define NUM_CLASSES  80
#define B_BATCH      8
#define M_BOX        100
#define M_PAD        112      // 7 tiles of 16
#define BOX_TILES    7
#define WAVES_PER_BLOCK 8

typedef __attribute__((ext_vector_type(2))) float v2f;
typedef __attribute__((ext_vector_type(8))) float v8f;

__global__ __launch_bounds__(256) void fcos_targets_wmma_kernel(
    const float* __restrict__ locations,        // L x 2
    const float* __restrict__ size_ranges,      // L x 2
    const float* __restrict__ strides_per_loc,  // L
    const float* __restrict__ bboxes,           // B x M x 4
    const int*   __restrict__ gt_classes,       // B x M
    float* __restrict__ out,                    // labels (B*L) then reg (B*L*4)
    int L, int numTiles)
{
    __shared__ float sx1[M_PAD], sy1[M_PAD], sx2[M_PAD], sy2[M_PAD];
    __shared__ int   scls[M_PAD];
    __shared__ float wA[WAVES_PER_BLOCK][16];
    __shared__ int   wI[WAVES_PER_BLOCK][16];

    const int tid = threadIdx.x;
    const int b   = blockIdx.y;

    // Stage this batch's boxes + classes in LDS (shared by all 8 waves).
    if (tid < M_PAD) {
        if (tid < M_BOX) {
            const float4 bb = ((const float4*)bboxes)[b * M_BOX + tid];
            sx1[tid] = bb.x; sy1[tid] = bb.y; sx2[tid] = bb.z; sy2[tid] = bb.w;
            scls[tid] = gt_classes[b * M_BOX + tid];
        } else {
            // Sentinel: l = xs - 1e9 < 0 -> never "in box" -> candidate INF,
            // and index >= M loses every INF tie to index 0. Never selected.
            sx1[tid] =  1.0e9f; sy1[tid] =  1.0e9f;
            sx2[tid] = -1.0e9f; sy2[tid] = -1.0e9f;
            scls[tid] = NUM_CLASSES;
        }
    }
    __syncthreads();

    const int wave = tid >> 5;
    const int lane = tid & 31;
    const int tile = blockIdx.x * WAVES_PER_BLOCK + wave;
    if (tile >= numTiles) return;        // wave-uniform: EXEC stays all-1s

    const int loc0   = tile * 16;
    const int col    = lane & 15;        // C/D column (box in tile); A row for low half
    const bool lowHalf = (lane < 16);

    // A matrix 16x4 f32 (ISA 7.12.2): lanes 0-15 -> (K0,K1)=(x,y) of M=lane,
    // lanes 16-31 -> (K2,K3)=(1,0).
    int aRow = loc0 + col; if (aRow >= L) aRow = L - 1;
    const float2 p = ((const float2*)locations)[aRow];
    const float xs = p.x, ys = p.y;
    v2f A;
    A.x = lowHalf ? xs : 1.0f;
    A.y = lowHalf ? ys : 0.0f;

    // Size ranges per C/D row slot r: row = loc0 + r (+8 for high half).
    float lo[8], hi[8];
#pragma unroll
    for (int r = 0; r < 8; ++r) {
        int row = loc0 + r + (lowHalf ? 0 : 8);
        if (row >= L) row = L - 1;
        const float2 sr = ((const float2*)size_ranges)[row];
        lo[r] = sr.x; hi[r] = sr.y;
    }

    float bestA[8]; int bestI[8];
#pragma unroll
    for (int r = 0; r < 8; ++r) { bestA[r] = 3.0e38f; bestI[r] = 0x7fffffff; }

    for (int tb = 0; tb < BOX_TILES; ++tb) {
        const int box = tb * 16 + col;
        const float x1 = sx1[box], y1 = sy1[box];
        const float x2 = sx2[box], y2 = sy2[box];
        const float areaBox = (x2 - x1) * (y2 - y1);

        // B matrix 4x16 f32: lane&15 = N; low half holds K0 (VGPR0) / K1
        // (VGPR1), high half holds K2 / K3 (mirrors the A-matrix K split).
        v2f Bl, Bt, Br, Bb;
        Bl.x = lowHalf ?  1.0f : -x1;  Bl.y = 0.0f;                  // l = xs - x1
        Bt.x = lowHalf ?  0.0f : -y1;  Bt.y = lowHalf ? 1.0f : 0.0f; // t = ys - y1
        Br.x = lowHalf ? -1.0f :  x2;  Br.y = 0.0f;                  // r = x2 - xs
        Bb.x = lowHalf ?  0.0f :  y2;  Bb.y = lowHalf ? -1.0f : 0.0f;// b = y2 - ys

        const v8f z = {};
        v8f Dl = __builtin_amdgcn_wmma_f32_16x16x4_f32(false, A, false, Bl, (short)0, z, false, false);
        v8f Dt = __builtin_amdgcn_wmma_f32_16x16x4_f32(false, A, false, Bt, (short)0, z, false, false);
        v8f Dr = __builtin_amdgcn_wmma_f32_16x16x4_f32(false, A, false, Br, (short)0, z, false, false);
        v8f Db = __builtin_amdgcn_wmma_f32_16x16x4_f32(false, A, false, Bb, (short)0, z, false, false);

#pragma unroll
        for (int r = 0; r < 8; ++r) {
            const float lv = Dl[r], tv = Dt[r], rv = Dr[r], bv = Db[r];
            const float mn = fminf(fminf(lv, tv), fminf(rv, bv));
            const float mx = fmaxf(fmaxf(lv, tv), fmaxf(rv, bv));
            // Branchless: bitwise & on comparisons, explicit selects.
            const int ok = (int)(mn > 0.0f) & (int)(mx >= lo[r]) & (int)(mx <= hi[r]);
            const float cand = ok ? areaBox : INF_VAL;
            // strict '<' keeps the earliest box index within this lane
            const int better = (int)(cand < bestA[r]) |
                               ((int)(cand == bestA[r]) & (int)(box < bestI[r]));
            bestA[r] = better ? cand : bestA[r];
            bestI[r] = better ? box  : bestI[r];
        }
    }

    // Min+argmin across the 16 lanes of each half (masks < 16 stay in-half).
#pragma unroll
    for (int r = 0; r < 8; ++r) {
#pragma unroll
        for (int m = 8; m >= 1; m >>= 1) {
            const float oA = __shfl_xor(bestA[r], m, 32);
            const int   oI = __shfl_xor(bestI[r], m, 32);
            const int better = (int)(oA < bestA[r]) |
                               ((int)(oA == bestA[r]) & (int)(oI < bestI[r]));
            bestA[r] = better ? oA : bestA[r];
            bestI[r] = better ? oI : bestI[r];
        }
    }

    // Transpose results to one-location-per-lane via per-wave LDS.
    // Same-wave DS ops are in-order (ISA 5, DScnt), no barrier needed.
    if (lane == 0) {
#pragma unroll
        for (int r = 0; r < 8; ++r) { wA[wave][r] = bestA[r]; wI[wave][r] = bestI[r]; }
    }
    if (lane == 16) {
#pragma unroll
        for (int r = 0; r < 8; ++r) { wA[wave][8 + r] = bestA[r]; wI[wave][8 + r] = bestI[r]; }
    }

    if (lane < 16) {
        const int loc = loc0 + lane;
        if (loc < L) {
            const float minA = wA[wave][lane];
            const int   idx  = wI[wave][lane];
            const float x1 = sx1[idx], y1 = sy1[idx];
            const float x2 = sx2[idx], y2 = sy2[idx];
            const float s  = strides_per_loc[loc];
            // xs,ys of this location were loaded by this lane in the A stage.
            float4 rt4;
            rt4.x = (xs - x1) / s;
            rt4.y = (ys - y1) / s;
            rt4.z = (x2 - xs) / s;
            rt4.w = (y2 - ys) / s;
            const int label = (minA >= INF_VAL) ? NUM_CLASSES : scls[idx];
            out[(size_t)b * L + loc] = (float)label;
            ((float4*)(out + (size_t)B_BATCH * L))[(size_t)b * L + loc] = rt4;
        }
    }
}

extern "C" void kernel_launch(void* const* d_in, const int* in_sizes, int n_in,
                              void* d_out, int out_size, void* d_ws, size_t ws_size,
                              hipStream_t stream) {
    (void)n_in; (void)out_size; (void)d_ws; (void)ws_size;
    const float* locations   = (const float*)d_in[0];   // L x 2
    const float* size_ranges = (const float*)d_in[1];   // L x 2
    const float* strides     = (const float*)d_in[2];   // L
    const float* bboxes      = (const float*)d_in[3];   // B x M x 4
    const int*   gt_classes  = (const int*)d_in[4];     // B x M
    float* out = (float*)d_out;   // labels (B*L) then reg targets (B*L*4)

    const int L = in_sizes[0] / 2;
    const int numTiles = (L + 15) / 16;
    dim3 grid((numTiles + WAVES_PER_BLOCK - 1) / WAVES_PER_BLOCK, B_BATCH);
    fcos_targets_wmma_kernel<<<grid, 256, 0, stream>>>(
        locations, size_ranges, strides, bboxes, gt_classes, out, L, numTiles);
}